// CondensedEmbracementLayer_69423851372962
// MI455X (gfx1250) — compile-verified
//
#include <hip/hip_runtime.h>
#include <stdint.h>

#define BS   64
#define SEQ  512
#define EMB  1024
#define NPAIR (BS * EMB / 2)   // 32768 threefry pairs
#define TPB  256               // threads per block (8 wave32 waves)
#define NBLK (NPAIR / TPB)     // 128 blocks

// gfx1250 async global->LDS DMA path (ASYNCcnt-tracked), guarded so the file
// compiles on toolchains where the builtins are absent (fallback: plain loads
// + global_prefetch_b8, which is probe-confirmed for gfx1250).
#if defined(__AMDGCN__) &&                                                     \
    __has_builtin(__builtin_amdgcn_global_load_async_to_lds_b32) &&            \
    __has_builtin(__builtin_amdgcn_s_wait_asynccnt)
#define CEL_ASYNC_LDS 1
#else
#define CEL_ASYNC_LDS 0
#endif

#if CEL_ASYNC_LDS
// Builtin params are (cuda_device int*, cuda_shared int*, imm, imm); target-AS
// annotated pointers convert since they map to the same physical AS (1 and 3).
typedef __attribute__((address_space(1))) int cel_gint_t;
typedef __attribute__((address_space(3))) int cel_lint_t;
#endif

__device__ __forceinline__ uint32_t cel_rotl32(uint32_t x, uint32_t r) {
  return (x << r) | (x >> (32u - r));
}

// threefry-2x32, key = (0, 42)  (jax.random.key(42)), exact JAX schedule:
// 5 groups of 4 rounds, rotation sets {13,15,26,6} / {17,29,16,24},
// key injection ks[(g+1)%3], ks[(g+2)%3]+g+1 after group g.
__device__ __forceinline__ void cel_threefry(uint32_t x0, uint32_t x1,
                                             uint32_t& o0, uint32_t& o1) {
  const uint32_t ks0 = 0u;
  const uint32_t ks1 = 42u;
  const uint32_t ks2 = 0x1BD11BDAu ^ 42u;  // ks0 ^ ks1 ^ parity
  x0 += ks0;
  x1 += ks1;
#define TF_R(R) { x0 += x1; x1 = cel_rotl32(x1, R); x1 ^= x0; }
  TF_R(13) TF_R(15) TF_R(26) TF_R(6)
  x0 += ks1; x1 += ks2 + 1u;
  TF_R(17) TF_R(29) TF_R(16) TF_R(24)
  x0 += ks2; x1 += ks0 + 2u;
  TF_R(13) TF_R(15) TF_R(26) TF_R(6)
  x0 += ks0; x1 += ks1 + 3u;
  TF_R(17) TF_R(29) TF_R(16) TF_R(24)
  x0 += ks1; x1 += ks2 + 4u;
  TF_R(13) TF_R(15) TF_R(26) TF_R(6)
  x0 += ks2; x1 += ks0 + 5u;
#undef TF_R
  o0 = x0;
  o1 = x1;
}

// bits -> uniform float in [0,1): bitcast((bits>>9) | 0x3f800000) - 1.0f
__device__ __forceinline__ float cel_u01(uint32_t bits) {
  return __uint_as_float((bits >> 9) | 0x3f800000u) - 1.0f;
}

__global__ __launch_bounds__(TPB) void
CondensedEmbracementLayer_69423851372962_kernel(const float* __restrict__ tok,
                                                const int* __restrict__ mask,
                                                float* __restrict__ out) {
  __shared__ int s_min[2];
#if CEL_ASYNC_LDS
  __shared__ int s_mask[2 * SEQ];
#endif

  const int t = threadIdx.x;
  const uint32_t i = (uint32_t)blockIdx.x * TPB + (uint32_t)t;  // pair index
  const int b0 = (int)(i >> 10);      // batch of output n0 = i   (const per block)
  const int b1 = b0 + 32;             // batch of output n1 = i + NPAIR
  const int j  = (int)(i & 1023u);    // embedding column for both outputs

  if (t < 2) s_min[t] = SEQ;

#if CEL_ASYNC_LDS
  {
    // Stage both mask rows into LDS with the async DMA path; the 20-round
    // threefry below executes entirely in the shadow of these transfers.
    int* g0 = const_cast<int*>(mask) + b0 * SEQ;
    int* g1 = const_cast<int*>(mask) + b1 * SEQ;
    __builtin_amdgcn_global_load_async_to_lds_b32(
        (cel_gint_t*)(g0 + t),        (cel_lint_t*)&s_mask[t],             0, 0);
    __builtin_amdgcn_global_load_async_to_lds_b32(
        (cel_gint_t*)(g0 + t + 256),  (cel_lint_t*)&s_mask[t + 256],       0, 0);
    __builtin_amdgcn_global_load_async_to_lds_b32(
        (cel_gint_t*)(g1 + t),        (cel_lint_t*)&s_mask[SEQ + t],       0, 0);
    __builtin_amdgcn_global_load_async_to_lds_b32(
        (cel_gint_t*)(g1 + t + 256),  (cel_lint_t*)&s_mask[SEQ + t + 256], 0, 0);
  }
#else
  // Fallback: prefetch mask rows (emits gfx1250 global_prefetch_b8).
  __builtin_prefetch(mask + b0 * SEQ + t, 0, 0);
  __builtin_prefetch(mask + b1 * SEQ + t, 0, 0);
#endif

  // --- pure-VALU PRNG work overlapped with the mask transfers ---
  uint32_t r0, r1;
  cel_threefry(i, i + (uint32_t)NPAIR, r0, r1);
  const float u0 = cel_u01(r0);
  const float u1 = cel_u01(r1);

  // --- leading-ones count = index of first zero (SEQ if none) ---
#if CEL_ASYNC_LDS
  __builtin_amdgcn_s_wait_asynccnt(0);  // my waves' DMA done
  __syncthreads();                       // everyone's DMA + s_min init visible
  {
    int m;
    m = s_mask[t];              if (m == 0) atomicMin(&s_min[0], t);
    m = s_mask[t + 256];        if (m == 0) atomicMin(&s_min[0], t + 256);
    m = s_mask[SEQ + t];        if (m == 0) atomicMin(&s_min[1], t);
    m = s_mask[SEQ + t + 256];  if (m == 0) atomicMin(&s_min[1], t + 256);
  }
#else
  __syncthreads();  // s_min init visible
  {
    int m;
    m = mask[b0 * SEQ + t];        if (m == 0) atomicMin(&s_min[0], t);
    m = mask[b0 * SEQ + t + 256];  if (m == 0) atomicMin(&s_min[0], t + 256);
    m = mask[b1 * SEQ + t];        if (m == 0) atomicMin(&s_min[1], t);
    m = mask[b1 * SEQ + t + 256];  if (m == 0) atomicMin(&s_min[1], t + 256);
  }
#endif
  __syncthreads();

  int c0 = s_min[0] - 1; if (c0 < 1) c0 = 1;   // L = max(count-1, 1)
  int c1 = s_min[1] - 1; if (c1 < 1) c1 = 1;
  const float L0 = (float)c0;
  const float L1 = (float)c1;

  int idx0 = (int)(u0 * L0); if (idx0 > SEQ - 1) idx0 = SEQ - 1;
  int idx1 = (int)(u1 * L1); if (idx1 > SEQ - 1) idx1 = SEQ - 1;

  // out[b, j] = tokens[b, idx, j]
  out[(uint32_t)b0 * EMB + (uint32_t)j] =
      tok[((uint32_t)b0 * SEQ + (uint32_t)idx0) * EMB + (uint32_t)j];
  out[(uint32_t)b1 * EMB + (uint32_t)j] =
      tok[((uint32_t)b1 * SEQ + (uint32_t)idx1) * EMB + (uint32_t)j];
}

extern "C" void kernel_launch(void* const* d_in, const int* in_sizes, int n_in,
                              void* d_out, int out_size, void* d_ws, size_t ws_size,
                              hipStream_t stream) {
  (void)in_sizes; (void)n_in; (void)out_size; (void)d_ws; (void)ws_size;
  const float* tok  = (const float*)d_in[0];  // [64, 512, 1024] f32
  const int*   mask = (const int*)d_in[1];    // [64, 512] i32
  float*       out  = (float*)d_out;          // [64, 1024] f32
  CondensedEmbracementLayer_69423851372962_kernel<<<dim3(NBLK), dim3(TPB), 0, stream>>>(
      tok, mask, out);
}